// PointNet2SemSegDitto_21895743275308
// MI455X (gfx1250) — compile-verified
//
#include <hip/hip_runtime.h>
#include <hip/hip_bf16.h>
#include <math.h>

#define B_ 8
#define N_ 8192

typedef float v2f __attribute__((ext_vector_type(2)));
typedef float v8f __attribute__((ext_vector_type(8)));

// ---------------------------------------------------------------------------
// Input transpose: xyz (B,6,N) -> pts (B,N,6) and coords (B,N,3)
// ---------------------------------------------------------------------------
__global__ __launch_bounds__(256) void k_transpose_in(
    const float* __restrict__ xyz, float* __restrict__ pts, float* __restrict__ coords) {
  int t = blockIdx.x * blockDim.x + threadIdx.x;
  if (t >= B_ * N_) return;
  int b = t / N_, n = t % N_;
  const float* src = xyz + (size_t)b * 6 * N_ + n;
  float c0 = src[0 * N_], c1 = src[1 * N_], c2 = src[2 * N_];
  float c3 = src[3 * N_], c4 = src[4 * N_], c5 = src[5 * N_];
  float* pr = pts + (size_t)t * 6;
  pr[0] = c0; pr[1] = c1; pr[2] = c2; pr[3] = c3; pr[4] = c4; pr[5] = c5;
  float* cr = coords + (size_t)t * 3;
  cr[0] = c0; cr[1] = c1; cr[2] = c2;
}

// ---------------------------------------------------------------------------
// Fold BN into scale/shift; pack weights in WMMA-B fragment order:
//   Wp[(kp*N + c)*2 + j] = W[(2*kp + j)*N + c]   (zero for rows >= K)
// so a lane's B fragment {row k+2h, row k+2h+1} is ONE aligned b64 load.
// ---------------------------------------------------------------------------
__global__ __launch_bounds__(256) void k_prep(
    const float* __restrict__ W, const float* __restrict__ bias,
    const float* __restrict__ g, const float* __restrict__ beta,
    const float* __restrict__ m, const float* __restrict__ v,
    float* __restrict__ Wp, float* __restrict__ scale, float* __restrict__ shift,
    int K, int Kpad, int N) {
  int t = blockIdx.x * blockDim.x + threadIdx.x;
  if (t < Kpad * N) {
    int k = t / N, c = t % N;
    int kp = k >> 1, j = k & 1;
    Wp[((size_t)kp * N + c) * 2 + j] = (k < K) ? W[t] : 0.0f;
  }
  if (t < N) {
    float sc = g[t] / sqrtf(v[t] + 1e-5f);
    scale[t] = sc;
    shift[t] = (bias[t] - m[t]) * sc + beta[t];
  }
}

// ---------------------------------------------------------------------------
// Farthest point sampling: one block per batch, dist[] + argmax tree in LDS.
// Records carry-in `far` each step (matches lax.scan output semantics);
// argmax tie-break prefers lowest index (matches jnp.argmax).
// ---------------------------------------------------------------------------
__global__ __launch_bounds__(1024) void k_fps(
    const float* __restrict__ xyz, int Nin, int npoint,
    int* __restrict__ out_idx, float* __restrict__ out_xyz) {
  __shared__ float dist[8192];
  __shared__ float sval[1024];
  __shared__ int   sidx[1024];
  __shared__ int   s_far;
  int b = blockIdx.x, t = threadIdx.x, nt = blockDim.x;
  const float* P = xyz + (size_t)b * Nin * 3;
  for (int i = t; i < Nin; i += nt) dist[i] = 1e10f;
  if (t == 0) s_far = 0;
  __syncthreads();
  for (int it = 0; it < npoint; ++it) {
    int far = s_far;
    if (t == 0) {
      out_idx[b * npoint + it] = far;
      float* o = out_xyz + (size_t)(b * npoint + it) * 3;
      o[0] = P[far * 3 + 0]; o[1] = P[far * 3 + 1]; o[2] = P[far * 3 + 2];
    }
    float cx = P[far * 3 + 0], cy = P[far * 3 + 1], cz = P[far * 3 + 2];
    float bv = -1.0f; int bi = 0;
    for (int i = t; i < Nin; i += nt) {
      float dx = P[i * 3 + 0] - cx, dy = P[i * 3 + 1] - cy, dz = P[i * 3 + 2] - cz;
      float d = dx * dx + dy * dy + dz * dz;
      float dn = dist[i];
      dn = d < dn ? d : dn;
      dist[i] = dn;
      if (dn > bv) { bv = dn; bi = i; }   // ascending i => lowest index kept on ties
    }
    sval[t] = bv; sidx[t] = bi;
    __syncthreads();
    for (int s = nt >> 1; s > 0; s >>= 1) {
      if (t < s) {
        float v2 = sval[t + s]; int i2 = sidx[t + s];
        if (v2 > sval[t] || (v2 == sval[t] && i2 < sidx[t])) { sval[t] = v2; sidx[t] = i2; }
      }
      __syncthreads();
    }
    if (t == 0) s_far = sidx[0];
    __syncthreads();
  }
}

// ---------------------------------------------------------------------------
// Ball query == first `nsample` indices (ascending) within radius, padded
// with the first hit (sort-of-masked-indices semantics of the reference).
// ---------------------------------------------------------------------------
__global__ __launch_bounds__(256) void k_ballquery(
    const float* __restrict__ xyz, const float* __restrict__ new_xyz,
    int* __restrict__ out, int Nin, int S, int nsample, float r2, int total) {
  int t = blockIdx.x * blockDim.x + threadIdx.x;
  if (t >= total) return;
  int b = t / S;
  const float* P = xyz + (size_t)b * Nin * 3;
  float qx = new_xyz[t * 3 + 0], qy = new_xyz[t * 3 + 1], qz = new_xyz[t * 3 + 2];
  int* o = out + (size_t)t * nsample;
  int cnt = 0, first = -1;
  for (int i = 0; i < Nin && cnt < nsample; ++i) {
    float dx = P[i * 3 + 0] - qx, dy = P[i * 3 + 1] - qy, dz = P[i * 3 + 2] - qz;
    float d = dx * dx + dy * dy + dz * dz;
    if (d <= r2) { if (first < 0) first = i; o[cnt++] = i; }
  }
  if (first < 0) first = Nin - 1;
  for (; cnt < nsample; ++cnt) o[cnt] = first;
}

// ---------------------------------------------------------------------------
// Grouping: X[row, :] = [xyz_i - new_xyz | feat_i | zero pad], row-major Kpad
// ---------------------------------------------------------------------------
__global__ __launch_bounds__(256) void k_group(
    const float* __restrict__ coords, const float* __restrict__ newxyz,
    const float* __restrict__ feat, const int* __restrict__ idx,
    float* __restrict__ out, int total, int S, int nsample, int featC, int Nin, int Kpad) {
  int r = blockIdx.x * blockDim.x + threadIdx.x;
  if (r >= total) return;
  int bs = r / nsample;
  int b = bs / S;
  int i = idx[r];
  const float* p = coords + ((size_t)b * Nin + i) * 3;
  const float* q = newxyz + (size_t)bs * 3;
  float* o = out + (size_t)r * Kpad;
  o[0] = p[0] - q[0]; o[1] = p[1] - q[1]; o[2] = p[2] - q[2];
  const float* f = feat + ((size_t)b * Nin + i) * featC;
  for (int c = 0; c < featC; ++c) o[3 + c] = f[c];
  for (int c = 3 + featC; c < Kpad; ++c) o[c] = 0.0f;
}

// ---------------------------------------------------------------------------
// Fused GEMM + BN + ReLU using V_WMMA_F32_16X16X4_F32 (full f32 precision).
// One wave computes a 32x32 output tile with 4 accumulators:
//   2 A-fragments (b64) + 2 packed-B fragments (b64) feed 4 WMMAs per K-step
//   => 1 VMEM op per WMMA, 2x operand reuse, and 4 independent acc chains
//      to cover the WMMA->WMMA RAW hazard window.
// A 16x4 f32 layout: lanes 0-15 hold K={k,k+1}, lanes 16-31 hold K={k+2,k+3}
// B  4x16 f32 layout: same K split, N striped across lane%16
// C/D: VGPR r -> (M = r + 8*(lane>=16), N = lane%16)
// Requires M % 32 == 0, N % 32 == 0, K % 4 == 0 (all true for this net).
// ---------------------------------------------------------------------------
__global__ __launch_bounds__(256) void k_gemm_bn_relu(
    const float* __restrict__ X, const float* __restrict__ Wp,
    const float* __restrict__ scale, const float* __restrict__ shift,
    float* __restrict__ Y, int M, int K, int N) {
  int wid  = blockIdx.x * (blockDim.x >> 5) + (threadIdx.x >> 5);
  int lane = threadIdx.x & 31;
  int tilesM = M >> 5;
  int tilesN = N >> 5;
  if (wid >= tilesM * tilesN) return;          // wave-uniform: EXEC stays all-1s
  int tm = wid % tilesM;
  int tn = wid / tilesM;
  int lm   = lane & 15;
  int half = lane >> 4;                        // 0 or 1
  int row0 = (tm << 5) + lm;                   // rows row0, row0+16
  int col0 = (tn << 5) + lm;                   // cols col0, col0+16
  int koff = half << 1;                        // 0 or 2
  const float* xr0 = X + (size_t)row0 * K + koff;
  const float* xr1 = xr0 + (size_t)16 * K;
  v8f a00 = {}, a01 = {}, a10 = {}, a11 = {};
  for (int k = 0; k < K; k += 4) {
    v2f af0 = *(const v2f*)(xr0 + k);          // A rows row0,   K pair
    v2f af1 = *(const v2f*)(xr1 + k);          // A rows row0+16
    size_t kp = (size_t)((k >> 1) + half) * N; // packed-B pair row
    v2f bf0 = *(const v2f*)(Wp + (kp + col0) * 2);
    v2f bf1 = *(const v2f*)(Wp + (kp + col0 + 16) * 2);
    a00 = __builtin_amdgcn_wmma_f32_16x16x4_f32(false, af0, false, bf0, (short)0, a00, false, false);
    a01 = __builtin_amdgcn_wmma_f32_16x16x4_f32(false, af0, false, bf1, (short)0, a01, false, false);
    a10 = __builtin_amdgcn_wmma_f32_16x16x4_f32(false, af1, false, bf0, (short)0, a10, false, false);
    a11 = __builtin_amdgcn_wmma_f32_16x16x4_f32(false, af1, false, bf1, (short)0, a11, false, false);
  }
  float sc0 = scale[col0],      sh0 = shift[col0];
  float sc1 = scale[col0 + 16], sh1 = shift[col0 + 16];
  size_t base = ((size_t)(tm << 5) + (half << 3)) * N;
#pragma unroll
  for (int r = 0; r < 8; ++r) {
    size_t ro = base + (size_t)r * N;
    float v00 = a00[r] * sc0 + sh0;
    float v01 = a01[r] * sc1 + sh1;
    float v10 = a10[r] * sc0 + sh0;
    float v11 = a11[r] * sc1 + sh1;
    Y[ro + col0]                       = v00 > 0.0f ? v00 : 0.0f;
    Y[ro + col0 + 16]                  = v01 > 0.0f ? v01 : 0.0f;
    Y[ro + (size_t)16 * N + col0]      = v10 > 0.0f ? v10 : 0.0f;
    Y[ro + (size_t)16 * N + col0 + 16] = v11 > 0.0f ? v11 : 0.0f;
  }
}

// ---------------------------------------------------------------------------
// Max over nsample axis: X (rows*nsample, C) -> Y (rows, C)
// ---------------------------------------------------------------------------
__global__ __launch_bounds__(256) void k_maxpool(
    const float* __restrict__ X, float* __restrict__ Y, int rows, int nsample, int C) {
  int t = blockIdx.x * blockDim.x + threadIdx.x;
  if (t >= rows * C) return;
  int c = t % C, r = t / C;
  const float* x = X + ((size_t)r * nsample) * C + c;
  float m = -3.4e38f;
  for (int j = 0; j < nsample; ++j) { float v = x[(size_t)j * C]; m = v > m ? v : m; }
  Y[t] = m;
}

// ---------------------------------------------------------------------------
// 3-NN + inverse-distance weights (normalized); tie-break = lowest index
// ---------------------------------------------------------------------------
__global__ __launch_bounds__(256) void k_knn3(
    const float* __restrict__ xyz1, const float* __restrict__ xyz2,
    int N1, int N2, int* __restrict__ idx3, float* __restrict__ w3, int total) {
  int t = blockIdx.x * blockDim.x + threadIdx.x;
  if (t >= total) return;
  int b = t / N1;
  float qx = xyz1[t * 3 + 0], qy = xyz1[t * 3 + 1], qz = xyz1[t * 3 + 2];
  const float* P = xyz2 + (size_t)b * N2 * 3;
  float d0 = 3.4e38f, d1 = 3.4e38f, d2 = 3.4e38f;
  int i0 = 0, i1 = 0, i2 = 0;
  for (int i = 0; i < N2; ++i) {
    float dx = P[i * 3 + 0] - qx, dy = P[i * 3 + 1] - qy, dz = P[i * 3 + 2] - qz;
    float d = dx * dx + dy * dy + dz * dz;
    if (d < d0)      { d2 = d1; i2 = i1; d1 = d0; i1 = i0; d0 = d; i0 = i; }
    else if (d < d1) { d2 = d1; i2 = i1; d1 = d;  i1 = i; }
    else if (d < d2) { d2 = d;  i2 = i; }
  }
  float w0 = 1.0f / (d0 + 1e-8f), w1 = 1.0f / (d1 + 1e-8f), w2 = 1.0f / (d2 + 1e-8f);
  float s = w0 + w1 + w2;
  w3[t * 3 + 0] = w0 / s; w3[t * 3 + 1] = w1 / s; w3[t * 3 + 2] = w2 / s;
  idx3[t * 3 + 0] = i0;   idx3[t * 3 + 1] = i1;   idx3[t * 3 + 2] = i2;
}

// ---------------------------------------------------------------------------
// Feature propagation input: out[row, :] = [pts1 (C1) | 3-NN interp (C2) | 0]
// ---------------------------------------------------------------------------
__global__ __launch_bounds__(256) void k_interp(
    const float* __restrict__ pts1, int C1, const float* __restrict__ pts2, int C2,
    const int* __restrict__ idx3, const float* __restrict__ w3,
    float* __restrict__ out, int N1, int N2, int Kpad, int total) {
  int t = blockIdx.x * blockDim.x + threadIdx.x;
  if (t >= total) return;
  int c = t % Kpad, r = t / Kpad;
  int b = r / N1;
  float v;
  if (c < C1) {
    v = pts1[(size_t)r * C1 + c];
  } else if (c < C1 + C2) {
    int cc = c - C1;
    const float* P2 = pts2 + (size_t)b * N2 * C2;
    v = w3[r * 3 + 0] * P2[(size_t)idx3[r * 3 + 0] * C2 + cc]
      + w3[r * 3 + 1] * P2[(size_t)idx3[r * 3 + 1] * C2 + cc]
      + w3[r * 3 + 2] * P2[(size_t)idx3[r * 3 + 2] * C2 + cc];
  } else {
    v = 0.0f;
  }
  out[t] = v;
}

// ---------------------------------------------------------------------------
// Head: sigmoid(x @ W2 + b2), one thread per point
// ---------------------------------------------------------------------------
__global__ __launch_bounds__(256) void k_head(
    const float* __restrict__ X, const float* __restrict__ W2,
    const float* __restrict__ b2, float* __restrict__ out, int total) {
  int t = blockIdx.x * blockDim.x + threadIdx.x;
  if (t >= total) return;
  const float* x = X + (size_t)t * 128;
  float s = b2[0];
  for (int c = 0; c < 128; ++c) s += x[c] * W2[c];
  out[t] = 1.0f / (1.0f + expf(-s));
}

// l2_pts (B,128,256) -> out (B,256,128)
__global__ __launch_bounds__(256) void k_out_l2(
    const float* __restrict__ l2, float* __restrict__ out, int total) {
  int t = blockIdx.x * blockDim.x + threadIdx.x;
  if (t >= total) return;
  int b = t / (128 * 256);
  int rem = t % (128 * 256);
  int s = rem / 256, c = rem % 256;
  out[(size_t)b * 256 * 128 + (size_t)c * 128 + s] = l2[t];
}

// ---------------------------------------------------------------------------
extern "C" void kernel_launch(void* const* d_in, const int* in_sizes, int n_in,
                              void* d_out, int out_size, void* d_ws, size_t ws_size,
                              hipStream_t stream) {
  (void)in_sizes; (void)n_in; (void)out_size; (void)ws_size;
  char* p = (char*)d_ws;
  auto alloc = [&](size_t nbytes) -> void* {
    void* r = (void*)p;
    p += (nbytes + 255) & ~(size_t)255;
    return r;
  };
  float* pts    = (float*)alloc(sizeof(float) * B_ * N_ * 6);
  float* coords = (float*)alloc(sizeof(float) * B_ * N_ * 3);
  float* l1_xyz = (float*)alloc(sizeof(float) * B_ * 512 * 3);
  float* l2_xyz = (float*)alloc(sizeof(float) * B_ * 128 * 3);
  float* l1_pts = (float*)alloc(sizeof(float) * B_ * 512 * 128);
  float* l2_pts = (float*)alloc(sizeof(float) * B_ * 128 * 256);
  float* l1_fp  = (float*)alloc(sizeof(float) * B_ * 512 * 256);
  float* l0_fp  = (float*)alloc(sizeof(float) * (size_t)B_ * N_ * 128);
  float* bufA   = (float*)alloc(sizeof(float) * 16777216);   // 64 MB ping
  float* bufB   = (float*)alloc(sizeof(float) * 16777216);   // 64 MB pong
  int*   fps1   = (int*)alloc(sizeof(int) * B_ * 512);
  int*   fps2   = (int*)alloc(sizeof(int) * B_ * 128);
  int*   idx1   = (int*)alloc(sizeof(int) * B_ * 512 * 32);
  int*   idx2   = (int*)alloc(sizeof(int) * B_ * 128 * 64);
  int*   knn2i  = (int*)alloc(sizeof(int) * B_ * 512 * 3);
  float* knn2w  = (float*)alloc(sizeof(float) * B_ * 512 * 3);
  int*   knn1i  = (int*)alloc(sizeof(int) * (size_t)B_ * N_ * 3);
  float* knn1w  = (float*)alloc(sizeof(float) * (size_t)B_ * N_ * 3);

  // 12 BN-GEMM layers: d_in base index, K, Kpad (mult of 4), Cout
  struct LD { int base, K, Kpad, N; };
  const LD L[12] = {
    { 1,   9,  12,  64}, { 7,  64,  64,  64}, {13,  64,  64, 128},   // sa1
    {19, 131, 132, 128}, {25, 128, 128, 128}, {31, 128, 128, 256},   // sa2
    {37, 384, 384, 512}, {43, 512, 512, 256},                        // fp2
    {49, 256, 256, 256}, {55, 256, 256, 128}, {61, 128, 128, 128},   // fp1
    {67, 128, 128, 128}                                              // head.c1
  };
  float* Wp[12]; float* Sc[12]; float* Sh[12];
  for (int i = 0; i < 12; ++i) {
    Wp[i] = (float*)alloc(sizeof(float) * L[i].Kpad * L[i].N);
    Sc[i] = (float*)alloc(sizeof(float) * L[i].N);
    Sh[i] = (float*)alloc(sizeof(float) * L[i].N);
    int total = L[i].Kpad * L[i].N;
    k_prep<<<(total + 255) / 256, 256, 0, stream>>>(
        (const float*)d_in[L[i].base + 0], (const float*)d_in[L[i].base + 1],
        (const float*)d_in[L[i].base + 2], (const float*)d_in[L[i].base + 3],
        (const float*)d_in[L[i].base + 4], (const float*)d_in[L[i].base + 5],
        Wp[i], Sc[i], Sh[i], L[i].K, L[i].Kpad, L[i].N);
  }

  auto gemm = [&](const float* X, float* Y, int li, int M) {
    int tiles = (M >> 5) * (L[li].N >> 5);
    k_gemm_bn_relu<<<(tiles + 7) / 8, 256, 0, stream>>>(
        X, Wp[li], Sc[li], Sh[li], Y, M, L[li].Kpad, L[li].N);
  };

  // ---- input transpose ----
  k_transpose_in<<<(B_ * N_ + 255) / 256, 256, 0, stream>>>((const float*)d_in[0], pts, coords);

  // ---- SA1 ----
  k_fps<<<B_, 1024, 0, stream>>>(coords, N_, 512, fps1, l1_xyz);
  k_ballquery<<<(B_ * 512 + 255) / 256, 256, 0, stream>>>(
      coords, l1_xyz, idx1, N_, 512, 32, 0.04f, B_ * 512);
  k_group<<<(B_ * 512 * 32 + 255) / 256, 256, 0, stream>>>(
      coords, l1_xyz, pts, idx1, bufA, B_ * 512 * 32, 512, 32, 6, N_, 12);
  gemm(bufA, bufB, 0, B_ * 512 * 32);
  gemm(bufB, bufA, 1, B_ * 512 * 32);
  gemm(bufA, bufB, 2, B_ * 512 * 32);
  k_maxpool<<<(B_ * 512 * 128 + 255) / 256, 256, 0, stream>>>(bufB, l1_pts, B_ * 512, 32, 128);

  // ---- SA2 ----
  k_fps<<<B_, 1024, 0, stream>>>(l1_xyz, 512, 128, fps2, l2_xyz);
  k_ballquery<<<(B_ * 128 + 255) / 256, 256, 0, stream>>>(
      l1_xyz, l2_xyz, idx2, 512, 128, 64, 0.16f, B_ * 128);
  k_group<<<(B_ * 128 * 64 + 255) / 256, 256, 0, stream>>>(
      l1_xyz, l2_xyz, l1_pts, idx2, bufA, B_ * 128 * 64, 128, 64, 128, 512, 132);
  gemm(bufA, bufB, 3, B_ * 128 * 64);
  gemm(bufB, bufA, 4, B_ * 128 * 64);
  gemm(bufA, bufB, 5, B_ * 128 * 64);
  k_maxpool<<<(B_ * 128 * 256 + 255) / 256, 256, 0, stream>>>(bufB, l2_pts, B_ * 128, 64, 256);

  // ---- FP2: l2 -> l1 ----
  k_knn3<<<(B_ * 512 + 255) / 256, 256, 0, stream>>>(
      l1_xyz, l2_xyz, 512, 128, knn2i, knn2w, B_ * 512);
  k_interp<<<(B_ * 512 * 384 + 255) / 256, 256, 0, stream>>>(
      l1_pts, 128, l2_pts, 256, knn2i, knn2w, bufA, 512, 128, 384, B_ * 512 * 384);
  gemm(bufA, bufB, 6, B_ * 512);
  gemm(bufB, l1_fp, 7, B_ * 512);

  // ---- FP1: l1 -> l0 ----
  k_knn3<<<(B_ * N_ + 255) / 256, 256, 0, stream>>>(
      coords, l1_xyz, N_, 512, knn1i, knn1w, B_ * N_);
  k_interp<<<(B_ * N_ * 256 + 255) / 256, 256, 0, stream>>>(
      nullptr, 0, l1_fp, 256, knn1i, knn1w, bufA, N_, 512, 256, B_ * N_ * 256);
  gemm(bufA, bufB, 8, B_ * N_);
  gemm(bufB, bufA, 9, B_ * N_);
  gemm(bufA, l0_fp, 10, B_ * N_);

  // ---- head ----
  gemm(l0_fp, bufA, 11, B_ * N_);
  k_head<<<(B_ * N_ + 255) / 256, 256, 0, stream>>>(
      bufA, (const float*)d_in[73], (const float*)d_in[74], (float*)d_out, B_ * N_);
  k_out_l2<<<(B_ * 128 * 256 + 255) / 256, 256, 0, stream>>>(
      l2_pts, (float*)d_out + B_ * N_, B_ * 128 * 256);
}